// GaussianKernel_76321568850191
// MI455X (gfx1250) — compile-verified
//
#include <hip/hip_runtime.h>

typedef __attribute__((ext_vector_type(2))) float v2f;
typedef __attribute__((ext_vector_type(8))) float v8f;

#define NPTS 1024
#define DDIM 128

static __device__ __forceinline__ v8f vzero8() {
    v8f v;
#pragma unroll
    for (int i = 0; i < 8; ++i) v[i] = 0.0f;
    return v;
}

// D = A(16x4 f32) * B(4x16 f32) + C(16x16 f32), wave32 WMMA
static __device__ __forceinline__ v8f wmma_f32(v2f a, v2f b, v8f c) {
    return __builtin_amdgcn_wmma_f32_16x16x4_f32(
        /*neg_a=*/false, a, /*neg_b=*/false, b,
        /*c_mod=*/(short)0, c, /*reuse_a=*/false, /*reuse_b=*/false);
}

// ws[0..1023] = ||X_i||^2 , ws[1024..2047] = ||Y_j||^2
__global__ __launch_bounds__(256) void gk_norms(const float* __restrict__ X,
                                                const float* __restrict__ Y,
                                                float* __restrict__ ws) {
    int tid = blockIdx.x * 256 + threadIdx.x;      // 0..2047
    const float* src = (tid < NPTS) ? X : Y;
    int row = tid & (NPTS - 1);
    const float4* p = (const float4*)(src + row * DDIM);
    float s = 0.0f;
#pragma unroll 8
    for (int i = 0; i < DDIM / 4; ++i) {
        float4 v = p[i];
        s += v.x * v.x + v.y * v.y + v.z * v.z + v.w * v.w;
    }
    ws[tid] = s;
}

// One wave handles a 16-row block of the output. Fused:
//   S = X Y^T  (WMMA f32)  ->  K = exp(..)  ->  rowsum(K), K@Y (WMMA f32)
__global__ __launch_bounds__(256) void gk_main(const float* __restrict__ X,
                                               const float* __restrict__ Y,
                                               const int* __restrict__ hp,
                                               float* __restrict__ Kout,
                                               float* __restrict__ dKout,
                                               const float* __restrict__ ws) {
    __shared__ float lds[8 * 16 * 18];             // per-wave 16x16 K tile, row stride 18
    const int lane = threadIdx.x & 31;
    const int wave = threadIdx.x >> 5;
    const int l15  = lane & 15;
    const int hi   = lane >> 4;                    // 0: low half, 1: high half
    const int m0   = (blockIdx.x * 8 + wave) * 16; // 16-row block of output
    float* myLds = lds + wave * (16 * 18);

    const float h = (float)hp[0];
    const float inv_h2 = 1.0f / (h * h);

    // A fragments of X (16x128), layout: A[m][k] at lane (m) / (m|16), vgpr pair k0+{0,1}|{2,3}
    v2f aX[32];
#pragma unroll
    for (int kk = 0; kk < 32; ++kk) {
        int k0 = kk * 4 + (hi ? 2 : 0);
        aX[kk] = *(const v2f*)(X + (m0 + l15) * DDIM + k0);
    }

    // xx for my 8 row slots (C/D layout rows: r + 8*hi)
    float xxv[8];
#pragma unroll
    for (int r = 0; r < 8; ++r) xxv[r] = ws[m0 + r + 8 * hi];

    v8f acc2[8];                                   // KY accumulator, 16x128
#pragma unroll
    for (int nt = 0; nt < 8; ++nt) acc2[nt] = vzero8();
    float racc[8];                                 // per-lane partial row sums
#pragma unroll
    for (int r = 0; r < 8; ++r) racc[r] = 0.0f;

    for (int jt = 0; jt < NPTS / 16; ++jt) {
        const int col0 = jt * 16;
        const float yyv = ws[NPTS + col0 + l15];

        if (jt < NPTS / 16 - 1)                    // prefetch next Y j-tile row
            __builtin_prefetch(Y + (col0 + 16 + l15) * DDIM, 0, 0);

        // ---- GEMM1: S = X_blk @ Y_jblk^T, K = 128 via 32 x WMMA f32 16x16x4
        v8f S = vzero8();
#pragma unroll
        for (int kk = 0; kk < 32; ++kk) {
            int k0 = kk * 4 + (hi ? 2 : 0);
            // B[k][n] = Y[col0+n][k] -> contiguous pair along k in Y's row
            v2f b = *(const v2f*)(Y + (col0 + l15) * DDIM + k0);
            S = wmma_f32(aX[kk], b, S);
        }

        // ---- exp, store K, stage tile to LDS (A layout source), row sums
        float kt[8];
#pragma unroll
        for (int r = 0; r < 8; ++r) {
            float d2 = fmaxf(xxv[r] + yyv - 2.0f * S[r], 0.0f);
            kt[r] = __expf(-0.5f * inv_h2 * d2);
            racc[r] += kt[r];
            Kout[(m0 + r + 8 * hi) * NPTS + col0 + l15] = kt[r];
            myLds[(r + 8 * hi) * 18 + l15] = kt[r];
        }
        __syncthreads();

        // ---- GEMM2: KY += K_tile(16x16) @ Y[col0:col0+16, :]  (4 k-steps x 8 n-tiles)
#pragma unroll
        for (int kk2 = 0; kk2 < 4; ++kk2) {
            int k0 = kk2 * 4 + (hi ? 2 : 0);
            v2f aK = *(const v2f*)(myLds + l15 * 18 + k0);  // A[m][k], m=l15
#pragma unroll
            for (int nt = 0; nt < 8; ++nt) {
                int krow = col0 + k0;
                v2f b2;
                b2.x = Y[krow * DDIM + nt * 16 + l15];
                b2.y = Y[(krow + 1) * DDIM + nt * 16 + l15];
                acc2[nt] = wmma_f32(aK, b2, acc2[nt]);
            }
        }
        __syncthreads();
    }

    // ---- row sums: reduce over the 16-lane half (rows match C/D layout)
    float sv[8];
#pragma unroll
    for (int r = 0; r < 8; ++r) {
        float s = racc[r];
        s += __shfl_xor(s, 1, 32);
        s += __shfl_xor(s, 2, 32);
        s += __shfl_xor(s, 4, 32);
        s += __shfl_xor(s, 8, 32);
        sv[r] = s;
    }

    // ---- d_K_sum = inv_h2 * (KY - s_i * X)
#pragma unroll
    for (int nt = 0; nt < 8; ++nt) {
#pragma unroll
        for (int r = 0; r < 8; ++r) {
            int m = m0 + r + 8 * hi;
            int n = nt * 16 + l15;
            float x = X[m * DDIM + n];
            dKout[m * DDIM + n] = inv_h2 * (acc2[nt][r] - sv[r] * x);
        }
    }
}

extern "C" void kernel_launch(void* const* d_in, const int* in_sizes, int n_in,
                              void* d_out, int out_size, void* d_ws, size_t ws_size,
                              hipStream_t stream) {
    const float* X = (const float*)d_in[0];
    const float* Y = (const float*)d_in[1];
    const int*   h = (const int*)d_in[2];
    float* out   = (float*)d_out;
    float* Kout  = out;                     // [1024,1024]
    float* dKout = out + NPTS * NPTS;       // [1024,128]
    float* ws    = (float*)d_ws;            // 2048 floats: xx | yy

    gk_norms<<<8, 256, 0, stream>>>(X, Y, ws);
    gk_main <<<8, 256, 0, stream>>>(X, Y, h, Kout, dKout, ws);
}